// AffinityPropagate3_34265249088049
// MI455X (gfx1250) — compile-verified
//
#include <hip/hip_runtime.h>
#include <cstdint>
#include <cstddef>

// ---------------- geometry ----------------
#define PT        16              // PROP_TIME
#define TILE      48              // output tile edge
#define RW        80              // region edge = TILE + 2*PT
#define RR        (RW * RW)       // 6400 region pixels
#define PAD0      128             // floats of guard space before arrays
#define KOFF      PAD0            // softmax-kernel planes (9 * RR)
#define AOFF      (KOFF + 9 * RR) // x buffer A
#define BOFF      (AOFF + RR)     // x buffer B (sparse depth lands here first)
#define TAILPAD   1312            // absorbs TDM shift-spill + halo reads
#define SMEM_FLOATS (BOFF + RR + TAILPAD)
#define SMEM_BYTES  (SMEM_FLOATS * 4)

static constexpr int IMG_H = 480;
static constexpr int IMG_W = 640;

typedef uint32_t u32x4 __attribute__((ext_vector_type(4)));
typedef int      i32x4 __attribute__((ext_vector_type(4)));
typedef int      i32x8 __attribute__((ext_vector_type(8)));

#if defined(__has_builtin)
#if __has_builtin(__builtin_amdgcn_tensor_load_to_lds) && __has_builtin(__builtin_amdgcn_s_wait_tensorcnt)
#define HAS_TDM 1
#endif
#endif
#ifndef HAS_TDM
#define HAS_TDM 0
#endif

#if HAS_TDM
// Build a Tensor DMA Descriptor (D#) per CDNA5 ISA
// and issue TENSOR_LOAD_TO_LDS. tile = RW x RW x planes contiguous in LDS.
// This toolchain exposes the 6-arg builtin:
//   (uint32x4 g0, int32x8 g1, int32x4 g2, int32x4 g3, int32x8 extra, i32 cpol)
__device__ __forceinline__ void tdm_load(uint32_t lds_bytes, uint64_t gaddr,
                                         uint32_t dim0, uint32_t dim1,
                                         uint32_t planes,
                                         uint64_t stride0, uint64_t stride1) {
  u32x4 g0;
  g0[0] = 1u;                                             // count=1, is_restore=0, no gather
  g0[1] = lds_bytes;                                      // lds_addr [63:32]
  g0[2] = (uint32_t)gaddr;                                // global_addr [95:64]
  g0[3] = ((uint32_t)(gaddr >> 32) & 0x01ffffffu)         // global_addr [120:96]
          | 0x80000000u;                                  // type=2 ("image") [127:126]
  i32x8 g1;
  g1[0] = (int)(2u << 16);                                // workgroup_mask=0, data_size=4B
  g1[1] = (int)((dim0 & 0xffffu) << 16);                  // tensor_dim0 lo -> bits 63:48
  g1[2] = (int)(((dim0 >> 16) & 0xffffu) | ((dim1 & 0xffffu) << 16)); // dim0 hi | dim1 lo
  g1[3] = (int)(((dim1 >> 16) & 0xffffu) | ((uint32_t)RW << 16));     // dim1 hi | tile_dim0
  g1[4] = (int)(((uint32_t)RW) | (planes << 16));         // tile_dim1 | tile_dim2
  g1[5] = (int)(uint32_t)stride0;                         // tensor_dim0_stride [31:0]
  g1[6] = (int)(((uint32_t)(stride0 >> 32) & 0xffffu)     // stride0 [47:32]
                | (((uint32_t)stride1 & 0xffffu) << 16)); // stride1 [15:0]
  g1[7] = (int)(uint32_t)(stride1 >> 16);                 // stride1 [47:16]
  i32x4 g2;
  g2[0] = (int)planes;                                    // tensor_dim2
  g2[1] = 1;                                              // tensor_dim3 (unused)
  g2[2] = 0;                                              // tensor_dim2_stride lo (unused)
  g2[3] = 0;                                              // stride hi | tile_dim3=0
  i32x4 g3 = {0, 0, 0, 0};                                // dims 4 unused
  i32x8 gx = {0, 0, 0, 0, 0, 0, 0, 0};                    // trailing group (unused)
  __builtin_amdgcn_tensor_load_to_lds(g0, g1, g2, g3, gx, 0);
}
#endif

#define ROWTAPS(KA, KB, KC, L, M, RV)                                          \
  {                                                                            \
    float4 kk;                                                                 \
    kk = *(const float4*)(Kb + (KA) * RR);                                     \
    ax = fmaf(kk.x, (L), ax);     ay = fmaf(kk.y, (M).x, ay);                  \
    az = fmaf(kk.z, (M).y, az);   aw = fmaf(kk.w, (M).z, aw);                  \
    kk = *(const float4*)(Kb + (KB) * RR);                                     \
    ax = fmaf(kk.x, (M).x, ax);   ay = fmaf(kk.y, (M).y, ay);                  \
    az = fmaf(kk.z, (M).z, az);   aw = fmaf(kk.w, (M).w, aw);                  \
    kk = *(const float4*)(Kb + (KC) * RR);                                     \
    ax = fmaf(kk.x, (M).y, ax);   ay = fmaf(kk.y, (M).z, ay);                  \
    az = fmaf(kk.z, (M).w, az);   aw = fmaf(kk.w, (RV), aw);                   \
  }

__global__ void __launch_bounds__(256)
affprop_kernel(const float* __restrict__ g_guided, const float* __restrict__ g_x,
               const float* __restrict__ g_sparse, float* __restrict__ g_out) {
  constexpr int H = IMG_H, W = IMG_W;
  extern __shared__ float smem[];
  float* sK = smem + KOFF;   // 9 softmax planes, RW-stride rows
  float* sA = smem + AOFF;   // x ping
  float* sB = smem + BOFF;   // x pong (sparse depth initially)

  const int tid = (int)threadIdx.x;
  const int ox0 = (int)blockIdx.x * TILE;
  const int oy0 = (int)blockIdx.y * TILE;
  const int b   = (int)blockIdx.z;

  const int gx0  = (ox0 - PT) < 0 ? 0 : (ox0 - PT);
  const int gy0  = (oy0 - PT) < 0 ? 0 : (oy0 - PT);
  const int scol = gx0 - (ox0 - PT);   // left clip shift (0..16)
  const int srow = gy0 - (oy0 - PT);   // top clip shift (0..16)

#if HAS_TDM
  // ---- async tile loads via Tensor Data Mover (wave 0 issues, all wait) ----
  if (tid < 32) {
    const uint32_t ldsbase = (uint32_t)(uintptr_t)smem;  // LDS byte offset
    const uint32_t shiftB  = (uint32_t)(srow * RW + scol) * 4u;
    const uint32_t d0 = (uint32_t)(W - gx0);             // right-edge zero fill
    const uint32_t d1 = (uint32_t)(H - gy0);             // bottom-edge zero fill
    tdm_load(ldsbase + (uint32_t)KOFF * 4u + shiftB,
             (uint64_t)(uintptr_t)(g_guided +
                 (((uint64_t)b * 9u) * H + (uint64_t)gy0) * W + gx0),
             d0, d1, 9u, (uint64_t)W, (uint64_t)H * (uint64_t)W);
    tdm_load(ldsbase + (uint32_t)AOFF * 4u + shiftB,
             (uint64_t)(uintptr_t)(g_x + ((uint64_t)b * H + gy0) * W + gx0),
             d0, d1, 1u, (uint64_t)W, 0u);
    tdm_load(ldsbase + (uint32_t)BOFF * 4u + shiftB,
             (uint64_t)(uintptr_t)(g_sparse + ((uint64_t)b * H + gy0) * W + gx0),
             d0, d1, 1u, (uint64_t)W, 0u);
    __builtin_amdgcn_s_wait_tensorcnt(0);
  }
  __syncthreads();
  // Zero the left/top out-of-image strips (TDM shift trick leaves spill there).
  if ((srow | scol) != 0) {
    for (int p = tid; p < RR; p += 256) {
      const int ry = p / RW, rx = p - ry * RW;
      if (ry < srow || rx < scol) {
#pragma unroll
        for (int z = 0; z < 9; ++z) sK[z * RR + p] = 0.0f;
        sA[p] = 0.0f;
        sB[p] = 0.0f;
      }
    }
    __syncthreads();
  }
#else
  // ---- fallback: plain guarded global->LDS copies ----
  for (int p = tid; p < RR; p += 256) {
    const int ry = p / RW, rx = p - ry * RW;
    const int gy = oy0 - PT + ry, gx = ox0 - PT + rx;
    const bool in = (gy >= 0) & (gy < H) & (gx >= 0) & (gx < W);
    const size_t gi = ((size_t)b * H + (size_t)gy) * W + (size_t)gx;
#pragma unroll
    for (int z = 0; z < 9; ++z)
      sK[z * RR + p] =
          in ? g_guided[(((size_t)b * 9 + z) * H + (size_t)gy) * W + gx] : 0.0f;
    sA[p] = in ? g_x[gi] : 0.0f;
    sB[p] = in ? g_sparse[gi] : 0.0f;
  }
  __syncthreads();
#endif

  // ---- per-pixel softmax over 9 taps; fold sparse mask into the kernel ----
  for (int p = tid; p < RR; p += 256) {
    float v[9];
    float m = -3.402823466e38f;
#pragma unroll
    for (int z = 0; z < 9; ++z) {
      v[z] = sK[z * RR + p];
      m = fmaxf(m, v[z]);
    }
    float s = 0.0f;
#pragma unroll
    for (int z = 0; z < 9; ++z) {
      v[z] = __expf(v[z] - m);
      s += v[z];
    }
    const float inv = 1.0f / s;
    const bool msk = sB[p] > 0.0f;  // sign(sparse_depth) in {0,1}
#pragma unroll
    for (int z = 0; z < 9; ++z)
      sK[z * RR + p] = msk ? (z == 4 ? 1.0f : 0.0f) : v[z] * inv;
  }
  __syncthreads();

  // ---- 16 Jacobi propagation steps entirely in LDS ----
  float* cur = sA;
  float* nxt = sB;
  for (int t = 0; t < PT; ++t) {
    for (int q = tid; q < RR / 4; q += 256) {
      const int ry = q / (RW / 4);
      const int c = (q - ry * (RW / 4)) << 2;
      const int base = ry * RW + c;
      const float4 m0 = *(const float4*)(cur + base - RW);
      const float  l0 = cur[base - RW - 1], r0 = cur[base - RW + 4];
      const float4 m1 = *(const float4*)(cur + base);
      const float  l1 = cur[base - 1],      r1 = cur[base + 4];
      const float4 m2 = *(const float4*)(cur + base + RW);
      const float  l2 = cur[base + RW - 1], r2 = cur[base + RW + 4];
      const float* Kb = sK + base;
      float ax = 0.0f, ay = 0.0f, az = 0.0f, aw = 0.0f;
      ROWTAPS(0, 1, 2, l0, m0, r0)
      ROWTAPS(3, 4, 5, l1, m1, r1)
      ROWTAPS(6, 7, 8, l2, m2, r2)
      float4 o;
      o.x = ax; o.y = ay; o.z = az; o.w = aw;
      *(float4*)(nxt + base) = o;
    }
    float* tsw = cur; cur = nxt; nxt = tsw;
    __syncthreads();
  }

  // ---- store central TILE x TILE (clipped) ----
  for (int q = tid; q < (TILE * TILE) / 4; q += 256) {
    const int iy = q / (TILE / 4);
    const int ix = (q - iy * (TILE / 4)) << 2;
    const int y = oy0 + iy, x = ox0 + ix;
    if (y < H && x < W) {
      const float4 v = *(const float4*)(cur + (iy + PT) * RW + (ix + PT));
      *(float4*)(g_out + ((size_t)b * H + y) * W + x) = v;
    }
  }
}

extern "C" void kernel_launch(void* const* d_in, const int* in_sizes, int n_in,
                              void* d_out, int out_size, void* d_ws, size_t ws_size,
                              hipStream_t stream) {
  (void)n_in; (void)d_ws; (void)ws_size; (void)out_size;
  const float* guided = (const float*)d_in[0];
  const float* x      = (const float*)d_in[1];
  const float* sparse = (const float*)d_in[2];
  float* out = (float*)d_out;

  const int B = in_sizes[1] / (IMG_H * IMG_W);

  static_assert(SMEM_BYTES <= 320 * 1024, "LDS budget exceeded");
  (void)hipFuncSetAttribute(reinterpret_cast<const void*>(affprop_kernel),
                            hipFuncAttributeMaxDynamicSharedMemorySize,
                            SMEM_BYTES);

  dim3 grid((IMG_W + TILE - 1) / TILE, (IMG_H + TILE - 1) / TILE, B);
  affprop_kernel<<<grid, 256, SMEM_BYTES, stream>>>(guided, x, sparse, out);
}